// maxF1_25838523253435
// MI455X (gfx1250) — compile-verified
//
#include <hip/hip_runtime.h>
#include <hip/hip_bf16.h>
#include <stdint.h>

#define NUM_T   1000
#define NBINS   1001
#define CLS     4
#define HW      (512 * 512)
#define CHUNK   16384              // elements per block (fits 16-bit packed counters)
#define TPB     256
#define TILE_FLOATS 1024           // TPB lanes * 4 floats (b128 per lane)
#define NTILES  (CHUNK / TILE_FLOATS)

// idx = #{k in [0,999] : t_k < x}, t_k = k/999  (searchsorted left).
// (int)(x*999) is within +/-1 of the true bin -> two branchless corrections.
__device__ __forceinline__ void hist_one(float x, float y, unsigned* hist) {
  const float step = 1.0f / 999.0f;
  int i = (int)fmaxf(fminf(x * 999.0f, 999.0f), 0.0f);
  i -= (i > 0     && (float)(i - 1) * step >= x) ? 1 : 0;
  i += (i < NUM_T && (float)i       * step <  x) ? 1 : 0;
  // packed: count in [31:16], y-sum in [15:0]; y is exactly 0.0f or 1.0f
  atomicAdd(&hist[i], 0x10000u + (unsigned)y);
}

__global__ __launch_bounds__(TPB) void maxf1_hist_kernel(
    const float* __restrict__ x, const float* __restrict__ y,
    unsigned* __restrict__ ys_g, unsigned* __restrict__ cnt_g) {
  __shared__ __align__(16) float xb[2][TILE_FLOATS];
  __shared__ __align__(16) float yb[2][TILE_FLOATS];
  __shared__ unsigned hist[NBINS];

  const unsigned tid = threadIdx.x;
  for (int i = tid; i < NBINS; i += TPB) hist[i] = 0u;
  __syncthreads();

  const int cls = (blockIdx.x / (HW / CHUNK)) & (CLS - 1); // layout [B,C,H,W]
  const size_t base = (size_t)blockIdx.x * CHUNK;
  const float* xp = x + base;
  const float* yp = y + base;

  // CDNA5 async global->LDS staging: 16B per lane, no dest VGPRs, ASYNCcnt tracked.
  auto issue = [&](int t, int slot) {
    unsigned ldsx = (unsigned)(uintptr_t)&xb[slot][tid * 4];
    unsigned ldsy = (unsigned)(uintptr_t)&yb[slot][tid * 4];
    unsigned long long gx =
        (unsigned long long)(uintptr_t)(xp + (size_t)t * TILE_FLOATS + tid * 4u);
    unsigned long long gy =
        (unsigned long long)(uintptr_t)(yp + (size_t)t * TILE_FLOATS + tid * 4u);
    asm volatile("global_load_async_to_lds_b128 %0, %1, off"
                 :: "v"(ldsx), "v"(gx) : "memory");
    asm volatile("global_load_async_to_lds_b128 %0, %1, off"
                 :: "v"(ldsy), "v"(gy) : "memory");
  };

  issue(0, 0);
  for (int t = 0; t < NTILES; ++t) {
    const int slot = t & 1;
    if (t + 1 < NTILES) {
      issue(t + 1, slot ^ 1);                         // 4 outstanding async ops
      asm volatile("s_wait_asynccnt 2" ::: "memory"); // tile t's 2 ops complete
    } else {
      asm volatile("s_wait_asynccnt 0" ::: "memory");
    }
    // each lane consumes exactly the bytes it async-loaded -> no barrier needed
    float4 xv = *(const float4*)&xb[slot][tid * 4];
    float4 yv = *(const float4*)&yb[slot][tid * 4];
    hist_one(xv.x, yv.x, hist);
    hist_one(xv.y, yv.y, hist);
    hist_one(xv.z, yv.z, hist);
    hist_one(xv.w, yv.w, hist);
  }

  __syncthreads();
  for (int i = tid; i < NBINS; i += TPB) {
    unsigned h = hist[i];
    if (h) {
      atomicAdd(&ys_g[cls * NBINS + i], h & 0xFFFFu);
      atomicAdd(&cnt_g[cls * NBINS + i], h >> 16);
    }
  }
}

__global__ __launch_bounds__(1024) void maxf1_reduce_kernel(
    const unsigned* __restrict__ ys_g, const unsigned* __restrict__ cnt_g,
    float* __restrict__ out, unsigned npc) {
  __shared__ unsigned scy[1024];
  __shared__ unsigned scc[1024];
  __shared__ float red[1024];
  const unsigned tid = threadIdx.x;
  float loss = 0.0f;

  for (int c = 0; c < CLS; ++c) {
    scy[tid] = (tid < NBINS) ? ys_g[c * NBINS + tid] : 0u;
    scc[tid] = (tid < NBINS) ? cnt_g[c * NBINS + tid] : 0u;
    __syncthreads();
    // inclusive scan (Hillis-Steele) over 1024 slots (1001 live bins)
    for (int off = 1; off < 1024; off <<= 1) {
      unsigned ay = (tid >= (unsigned)off) ? scy[tid - off] : 0u;
      unsigned ac = (tid >= (unsigned)off) ? scc[tid - off] : 0u;
      __syncthreads();
      scy[tid] += ay;
      scc[tid] += ac;
      __syncthreads();
    }
    const unsigned total_y = scy[NBINS - 1];   // sum over all idx 0..1000
    float f1 = 0.0f;
    if (tid < NUM_T) {
      float tp = (float)(total_y - scy[tid]);  // counts <= 2^22 -> exact in f32
      float pp = (float)(npc - scc[tid]);
      float fp = pp - tp;
      float fn = (float)total_y - tp;
      float denom = tp + 0.5f * (fn + fp);
      f1 = (denom == 0.0f) ? 0.0f : tp / denom;
    }
    red[tid] = f1;
    __syncthreads();
    for (int off = 512; off > 0; off >>= 1) {
      if (tid < (unsigned)off) red[tid] = fmaxf(red[tid], red[tid + off]);
      __syncthreads();
    }
    loss += 1.0f - red[0];
    __syncthreads();
  }
  if (tid == 0) out[0] = loss * (1.0f / CLS);
}

extern "C" void kernel_launch(void* const* d_in, const int* in_sizes, int n_in,
                              void* d_out, int out_size, void* d_ws, size_t ws_size,
                              hipStream_t stream) {
  const float* x = (const float*)d_in[0];
  const float* y = (const float*)d_in[1];
  float* out = (float*)d_out;

  unsigned* ys_g  = (unsigned*)d_ws;           // [CLS][NBINS] y-sum accumulators
  unsigned* cnt_g = ys_g + CLS * NBINS;        // [CLS][NBINS] count accumulators

  // deterministic: zero accumulators every call (graph-capture-safe memset node)
  hipMemsetAsync(d_ws, 0, 2ull * CLS * NBINS * sizeof(unsigned), stream);

  const int ntot = in_sizes[0];                // B*C*H*W = 16,777,216
  const int blocks = ntot / CHUNK;             // 1024 blocks
  maxf1_hist_kernel<<<blocks, TPB, 0, stream>>>(x, y, ys_g, cnt_g);
  maxf1_reduce_kernel<<<1, 1024, 0, stream>>>(ys_g, cnt_g, out,
                                              (unsigned)(ntot / CLS));
}